// Frobenius_71975061946522
// MI455X (gfx1250) — compile-verified
//
#include <hip/hip_runtime.h>
#include <hip/hip_bf16.h>

#define NMAT  4096
#define TILE  128
#define NBX   (NMAT / TILE)      // 32 blocks per dimension
#define NBLK  (NBX * NBX)        // 1024 blocks
#define LDSTR 132                // padded LDS row stride (floats): 4-bank advance/row
#define ITERS 20

typedef __attribute__((ext_vector_type(2))) float v2f;
typedef __attribute__((ext_vector_type(8))) float v8f;

__global__ void fro_init(int* doneFlag) { *doneFlag = 0; }

// MODE 0: sums-only pass over src (bootstrap row/col/total of X0)
// MODE 1: fused projection update: dst = relu(P1(src)), plus partial sums of
//         the NEW matrix (for next iteration) and block-min of pre-relu values.
template <int MODE>
__global__ __launch_bounds__(256) void fro_pass(
    const float* __restrict__ src, float* __restrict__ dst,
    const float* __restrict__ rowSum, const float* __restrict__ colSum,
    const float* __restrict__ scalars,
    float* __restrict__ rowPart, float* __restrict__ colPart,
    float* __restrict__ blockMin, const int* __restrict__ doneFlag)
{
    if (MODE == 1) {
        if (*doneFlag) return;          // uniform: freeze X once converged
    }

    __shared__ __align__(16) float tile[TILE][LDSTR];   // updated 128x128 tile
    __shared__ __align__(16) float colAccLds[8][TILE];
    __shared__ float minW[8];

    const int t    = threadIdx.x;
    const int lane = t & 31;
    const int w    = t >> 5;            // wave 0..7, owns 16 rows of the tile
    const int bx   = blockIdx.x;        // column block
    const int by   = blockIdx.y;        // row block
    const int col0 = bx * TILE + lane * 4;
    const int row0 = by * TILE + w * 16;
    const float invn = 1.0f / (float)NMAT;

    float k1 = 0.0f;
    float cax = 0.f, cay = 0.f, caz = 0.f, caw = 0.f;
    if (MODE == 1) {
        k1  = scalars[0];               // 1/n + total/n^2 (from finalize)
        cax = colSum[col0 + 0] * invn;
        cay = colSum[col0 + 1] * invn;
        caz = colSum[col0 + 2] * invn;
        caw = colSum[col0 + 3] * invn;
    }

    float sx = 0.f, sy = 0.f, sz = 0.f, sw = 0.f;   // per-thread column sums
    float mn = 3.402823466e38f;

    // ---- Phase 1: coalesced stream (load -> update -> store + LDS stage) ----
    for (int r = 0; r < 16; ++r) {
        const int row = row0 + r;
        const float4 v =
            *reinterpret_cast<const float4*>(src + (size_t)row * NMAT + col0);
        float4 o;
        if (MODE == 1) {
            const float ra = rowSum[row] * invn;    // wave-uniform
            const float px = v.x + k1 - ra - cax;
            const float py = v.y + k1 - ra - cay;
            const float pz = v.z + k1 - ra - caz;
            const float pw = v.w + k1 - ra - caw;
            mn = fminf(mn, fminf(fminf(px, py), fminf(pz, pw)));
            o.x = fmaxf(px, 0.0f); o.y = fmaxf(py, 0.0f);
            o.z = fmaxf(pz, 0.0f); o.w = fmaxf(pw, 0.0f);
            *reinterpret_cast<float4*>(dst + (size_t)row * NMAT + col0) = o;
            if (row + 8 < NMAT)   // uniform guard; emits global_prefetch_b8
                __builtin_prefetch(src + (size_t)(row + 8) * NMAT + col0, 0, 0);
        } else {
            o = v;
        }
        sx += o.x; sy += o.y; sz += o.z; sw += o.w;
        *reinterpret_cast<float4*>(&tile[w * 16 + r][lane * 4]) = o;
    }

    {   // stage per-thread column sums for cross-wave (deterministic) combine
        float* cp = &colAccLds[w][lane * 4];
        cp[0] = sx; cp[1] = sy; cp[2] = sz; cp[3] = sw;
    }
    if (MODE == 1) {
        #pragma unroll
        for (int i = 16; i >= 1; i >>= 1) mn = fminf(mn, __shfl_xor(mn, i, 32));
        if (lane == 0) minW[w] = mn;
    }
    __syncthreads();

    if (t < TILE) {
        float a = 0.0f;
        #pragma unroll
        for (int k = 0; k < 8; ++k) a += colAccLds[k][t];
        colPart[(size_t)by * NMAT + bx * TILE + t] = a;
    }
    if (MODE == 1 && t == 0) {
        float m2 = minW[0];
        #pragma unroll
        for (int k = 1; k < 8; ++k) m2 = fminf(m2, minW[k]);
        blockMin[by * NBX + bx] = m2;
    }

    // ---- Phase 2: WMMA row sums of this wave's updated 16x128 strip ----
    // A = 16x4 data chunk (lane<16: row=lane, K=0..1; lane>=16: row=lane-16,
    // K=2..3), B = all-ones -> D[m][*] = row sum, accumulated over 32 chunks.
    const int half = lane >> 4;
    const int mrow = lane & 15;
    const float* lrow = &tile[w * 16 + mrow][0];
    v2f ones; ones[0] = 1.0f; ones[1] = 1.0f;
    v8f acc = {0.f, 0.f, 0.f, 0.f, 0.f, 0.f, 0.f, 0.f};
    #pragma unroll 4
    for (int c0 = 0; c0 < TILE; c0 += 4) {
        v2f a;
        a[0] = lrow[c0 + half * 2 + 0];
        a[1] = lrow[c0 + half * 2 + 1];
        acc = __builtin_amdgcn_wmma_f32_16x16x4_f32(false, a, false, ones,
                                                    (short)0, acc, false, false);
    }
    // C/D layout: VGPR i -> (M=i, N=lane) for lanes 0-15, (M=i+8, N=lane-16)
    // for lanes 16-31. Lane 0 holds rows 0..7, lane 16 holds rows 8..15 (N=0).
    if (mrow == 0) {
        const int rb = by * TILE + w * 16 + half * 8;
        #pragma unroll
        for (int i = 0; i < 8; ++i)
            rowPart[(size_t)bx * NMAT + rb + i] = acc[i];
    }
}

// Single-block deterministic fold: partials -> rowSum/colSum/total (+k1),
// block mins -> global min -> done flag. FIRST_PASS skips the min/flag part.
template <int FIRST_PASS>
__global__ __launch_bounds__(1024) void fro_finalize(
    float* __restrict__ rowSum, float* __restrict__ colSum,
    float* __restrict__ scalars,
    const float* __restrict__ rowPart, const float* __restrict__ colPart,
    const float* __restrict__ blockMin, int* __restrict__ doneFlag)
{
    __shared__ float s[1024];
    const int t = threadIdx.x;
    if (!FIRST_PASS) {
        if (*doneFlag) return;
        s[t] = blockMin[t];             // exactly NBLK == 1024 entries
        __syncthreads();
        for (int st = 512; st > 0; st >>= 1) {
            if (t < st) s[t] = fminf(s[t], s[t + st]);
            __syncthreads();
        }
        const float gmin = s[0];
        __syncthreads();
        if (gmin >= 0.0f) {             // converged: freeze from next iter on
            if (t == 0) *doneFlag = 1;
            return;
        }
    }
    float tot = 0.0f;
    #pragma unroll
    for (int k = 0; k < NMAT / 1024; ++k) {
        const int r = t + k * 1024;
        float a = 0.0f, c = 0.0f;
        for (int b = 0; b < NBX; ++b) {
            a += rowPart[(size_t)b * NMAT + r];
            c += colPart[(size_t)b * NMAT + r];
        }
        rowSum[r] = a; colSum[r] = c; tot += a;
    }
    s[t] = tot;
    __syncthreads();
    for (int st = 512; st > 0; st >>= 1) {
        if (t < st) s[t] += s[t + st];
        __syncthreads();
    }
    if (t == 0)
        scalars[0] = 1.0f / (float)NMAT + s[0] / ((float)NMAT * (float)NMAT);
}

extern "C" void kernel_launch(void* const* d_in, const int* in_sizes, int n_in,
                              void* d_out, int out_size, void* d_ws, size_t ws_size,
                              hipStream_t stream)
{
    const float* X = (const float*)d_in[0];   // 4096x4096 fp32
    float* out = (float*)d_out;

    float* w        = (float*)d_ws;
    float* rowSum   = w;  w += NMAT;
    float* colSum   = w;  w += NMAT;
    float* rowPart  = w;  w += (size_t)NBX * NMAT;   // [colBlock][row]
    float* colPart  = w;  w += (size_t)NBX * NMAT;   // [rowBlock][col]
    float* blockMin = w;  w += NBLK;
    float* scalars  = w;  w += 8;
    int*   doneFlag = (int*)w;

    dim3 grid(NBX, NBX), blk(256);

    fro_init<<<1, 1, 0, stream>>>(doneFlag);
    // Bootstrap: sums of X0, then 20 fused update iterations (fixed launch
    // sequence; data-dependent early exit realized via device-side done flag).
    fro_pass<0><<<grid, blk, 0, stream>>>(X, out, rowSum, colSum, scalars,
                                          rowPart, colPart, blockMin, doneFlag);
    fro_finalize<1><<<1, 1024, 0, stream>>>(rowSum, colSum, scalars,
                                            rowPart, colPart, blockMin, doneFlag);
    for (int it = 0; it < ITERS; ++it) {
        const float* s = (it == 0) ? X : out;
        fro_pass<1><<<grid, blk, 0, stream>>>(s, out, rowSum, colSum, scalars,
                                              rowPart, colPart, blockMin, doneFlag);
        if (it + 1 < ITERS)
            fro_finalize<0><<<1, 1024, 0, stream>>>(rowSum, colSum, scalars,
                                                    rowPart, colPart, blockMin,
                                                    doneFlag);
    }
}